// BandpassFilterLayer_89421219102902
// MI455X (gfx1250) — compile-verified
//
#include <hip/hip_runtime.h>

#define BATCH 1024
#define TT 16384
#define PAD 27
#define ELEN (TT + 2 * PAD)        // 16438
#define NBLK ((ELEN + 15) / 16)    // 1028 blocks of 16 samples
#define EDGE_STRIDE 64             // per-row edge storage (2*PAD=54 used)
#define LDS_STRIDE 20              // 16-float rows padded to 20 words: 16B aligned, conflict-free

typedef float v2f __attribute__((ext_vector_type(2)));
typedef float v4f __attribute__((ext_vector_type(4)));
typedef float v8f __attribute__((ext_vector_type(8)));
typedef unsigned int u32x4 __attribute__((ext_vector_type(4)));
typedef int i32x4 __attribute__((ext_vector_type(4)));
typedef int i32x8 __attribute__((ext_vector_type(8)));

#if defined(__has_builtin)
#if __has_builtin(__builtin_amdgcn_tensor_load_to_lds) && \
    __has_builtin(__builtin_amdgcn_s_wait_tensorcnt)
#define HAVE_TDM 1
#endif
#endif
#ifndef HAVE_TDM
#define HAVE_TDM 0
#endif

// D = A(16x4) * B(4x16) + C, f32 WMMA (wave32)
__device__ __forceinline__ v8f wmma4(v2f A, v2f B, v8f C) {
  return __builtin_amdgcn_wmma_f32_16x16x4_f32(false, A, false, B, (short)0, C,
                                               false, false);
}

// ---------------------------------------------------------------------------
// Kernel 1: build block-filter matrices in f64 from runtime (b, a).
//   State space (DF2T, 8 states): s_t = A s_{t-1} + Bv x_t ; y_t = s_{t-1}[0] + b0 x_t
//   Tm[m][j] = b0 (j==m) | C A^{m-1-j} Bv (j<m)    (16x16 lower triangular)
//   Hm[m][k] = (A^m)[0][k]                          (16x16, cols 8..15 zero)
//   A16      = A^16 zero-padded to 16x16
//   Bb[i][j] = (A^{15-j} Bv)[i] zero-padded to 16x16
// Layout in mats[]: Tm@0, Hm@256, A16@512, Bb@768 (row-major float)
// ---------------------------------------------------------------------------
__global__ void gen_mats_kernel(const float* __restrict__ b,
                                const float* __restrict__ a,
                                float* __restrict__ mats) {
  if (threadIdx.x != 0 || blockIdx.x != 0) return;
  double bb[9], aa[9];
  for (int i = 0; i < 9; ++i) { bb[i] = (double)b[i]; aa[i] = (double)a[i]; }
  const double d0 = bb[0];
  double Bv[8];
  double A[8][8];
  for (int i = 0; i < 8; ++i)
    for (int k = 0; k < 8; ++k) A[i][k] = 0.0;
  for (int i = 0; i < 8; ++i) {
    Bv[i] = bb[i + 1] - aa[i + 1] * d0;
    A[i][0] = -aa[i + 1];
    if (i < 7) A[i][i + 1] = 1.0;
  }
  double Tm[16][16], Hm[16][16], A16[16][16], Bb[16][16];
  for (int m = 0; m < 16; ++m)
    for (int k = 0; k < 16; ++k) {
      Tm[m][k] = 0.0; Hm[m][k] = 0.0; A16[m][k] = 0.0; Bb[m][k] = 0.0;
    }
  for (int m = 0; m < 16; ++m) Tm[m][m] = d0;

  double Ap[8][8];  // A^p, starts at identity
  for (int i = 0; i < 8; ++i)
    for (int k = 0; k < 8; ++k) Ap[i][k] = (i == k) ? 1.0 : 0.0;

  for (int p = 0; p < 16; ++p) {
    for (int k = 0; k < 8; ++k) Hm[p][k] = Ap[0][k];      // C A^p
    double hp = 0.0;
    for (int k = 0; k < 8; ++k) hp += Ap[0][k] * Bv[k];   // C A^p Bv
    for (int j = 0; j + 1 + p < 16; ++j) Tm[j + 1 + p][j] = hp;
    for (int i = 0; i < 8; ++i) {                          // (A^p Bv)
      double s = 0.0;
      for (int k = 0; k < 8; ++k) s += Ap[i][k] * Bv[k];
      Bb[i][15 - p] = s;
    }
    double An[8][8];                                       // Ap = A * Ap
    for (int i = 0; i < 8; ++i)
      for (int k = 0; k < 8; ++k) {
        double s = 0.0;
        for (int q = 0; q < 8; ++q) s += A[i][q] * Ap[q][k];
        An[i][k] = s;
      }
    for (int i = 0; i < 8; ++i)
      for (int k = 0; k < 8; ++k) Ap[i][k] = An[i][k];
  }
  for (int i = 0; i < 8; ++i)
    for (int k = 0; k < 8; ++k) A16[i][k] = Ap[i][k];      // A^16

  for (int m = 0; m < 16; ++m)
    for (int k = 0; k < 16; ++k) {
      mats[0   + m * 16 + k] = (float)Tm[m][k];
      mats[256 + m * 16 + k] = (float)Hm[m][k];
      mats[512 + m * 16 + k] = (float)A16[m][k];
      mats[768 + m * 16 + k] = (float)Bb[m][k];
    }
}

// ---------------------------------------------------------------------------
// Extended-signal access helpers (odd padding, filtfilt edge handling)
// ---------------------------------------------------------------------------
__device__ __forceinline__ float read_fwd(const float* __restrict__ x,
                                          long rbase, int i) {
  if (i < PAD) return 2.0f * x[rbase] - x[rbase + (PAD - i)];
  if (i < PAD + TT) return x[rbase + i - PAD];
  if (i < ELEN) {
    int j = i - PAD - TT;
    return 2.0f * x[rbase + TT - 1] - x[rbase + TT - 2 - j];
  }
  return 0.0f;
}

__device__ __forceinline__ void write_fwd(float* __restrict__ out,
                                          float* __restrict__ edge, long rbase,
                                          long redge, int i, float v) {
  if (i >= PAD && i < PAD + TT) out[rbase + i - PAD] = v;
  else if (i < PAD) edge[redge + i] = v;
  else if (i < ELEN) edge[redge + i - TT] = v;
}

__device__ __forceinline__ float read_bwd(const float* __restrict__ out,
                                          const float* __restrict__ edge,
                                          long rbase, long redge, int i) {
  if (i < 0) return 0.0f;
  if (i < PAD) return edge[redge + i];
  if (i < PAD + TT) return out[rbase + i - PAD];
  if (i < ELEN) return edge[redge + i - TT];
  return 0.0f;
}

// ---------------------------------------------------------------------------
// Kernel 2: blocked filtfilt. One wave (32 threads) per 16-row tile.
// Per 16-sample block:  Y = Tm@X + Hm@S ;  S' = A16@S + Bb@X   (12 WMMAs)
// Bulk forward-pass X tiles are DMA'd into LDS by the Tensor Data Mover
// (double-buffered, TENSORcnt-tracked); edges use a guarded scalar path.
// ---------------------------------------------------------------------------
__global__ __launch_bounds__(32) void filtfilt_wmma_kernel(
    const float* __restrict__ x, const float* __restrict__ zi,
    const float* __restrict__ mats, float* __restrict__ edge,
    float* __restrict__ out) {
  __shared__ __align__(16) float xtA[16 * LDS_STRIDE];
  __shared__ __align__(16) float xtB[16 * LDS_STRIDE];
  __shared__ __align__(16) float st[16 * LDS_STRIDE];

  const int lane = threadIdx.x;
  const int hi = lane >> 4;   // 16-lane half
  const int ml = lane & 15;   // row-in-tile (M for A-operands, N for B/C/D)
  const int r = blockIdx.x * 16 + ml;
  const long rbase = (long)r * TT;
  const long redge = (long)r * EDGE_STRIDE;
  const size_t tilebase = (size_t)blockIdx.x * 16 * (size_t)TT;  // tile row 0

  // Preload A-operands in WMMA A-layout: lane(ml)=M, K = 4*kk + 2*hi + {0,1}
  v2f Tm[4], Bb[4], Hm[2], A16[2];
#pragma unroll
  for (int kk = 0; kk < 4; ++kk) {
    int c = 4 * kk + 2 * hi;
    Tm[kk].x = mats[0 + ml * 16 + c];     Tm[kk].y = mats[0 + ml * 16 + c + 1];
    Bb[kk].x = mats[768 + ml * 16 + c];   Bb[kk].y = mats[768 + ml * 16 + c + 1];
    if (kk < 2) {
      Hm[kk].x  = mats[256 + ml * 16 + c]; Hm[kk].y  = mats[256 + ml * 16 + c + 1];
      A16[kk].x = mats[512 + ml * 16 + c]; A16[kk].y = mats[512 + ml * 16 + c + 1];
    }
  }
  const float zi_a = zi[2 * hi], zi_b = zi[2 * hi + 1];
  const float zi_c = zi[4 + 2 * hi], zi_d = zi[5 + 2 * hi];

  v2f SB[2];  // running state, WMMA B-layout (K=state, N=row)

  auto lds_read_xb = [&](const float* buf, v2f XB[4]) {
#pragma unroll
    for (int kk = 0; kk < 4; ++kk)
      XB[kk] = *(const v2f*)&buf[ml * LDS_STRIDE + 4 * kk + 2 * hi];
  };

  // 12 WMMAs + S' D->B relayout through LDS; returns Y (D-layout)
  auto do_step = [&](const v2f XB[4]) -> v8f {
    v8f y = {};
#pragma unroll
    for (int kk = 0; kk < 4; ++kk) y = wmma4(Tm[kk], XB[kk], y);
#pragma unroll
    for (int kk = 0; kk < 2; ++kk) y = wmma4(Hm[kk], SB[kk], y);
    v8f sn = {};
#pragma unroll
    for (int kk = 0; kk < 2; ++kk) sn = wmma4(A16[kk], SB[kk], sn);
#pragma unroll
    for (int kk = 0; kk < 4; ++kk) sn = wmma4(Bb[kk], XB[kk], sn);
    v4f slo = {sn[0], sn[1], sn[2], sn[3]};
    v4f shi_ = {sn[4], sn[5], sn[6], sn[7]};
    *(v4f*)&st[ml * LDS_STRIDE + 8 * hi] = slo;
    *(v4f*)&st[ml * LDS_STRIDE + 8 * hi + 4] = shi_;
    __syncthreads();
    SB[0] = *(v2f*)&st[ml * LDS_STRIDE + 2 * hi];
    SB[1] = *(v2f*)&st[ml * LDS_STRIDE + 4 + 2 * hi];
    return y;
  };

  // Guarded path for the 5 edge blocks per pass
  auto edge_block = [&](int dir, int blk) {
    float xv[8];
#pragma unroll
    for (int q = 0; q < 8; ++q) {
      const int m = 8 * hi + q;
      const int j = 16 * blk + m;
      const int i = dir ? (ELEN - 1 - j) : j;
      xv[q] = dir ? read_bwd(out, edge, rbase, redge, i)
                  : read_fwd(x, rbase, i);
    }
    v4f a = {xv[0], xv[1], xv[2], xv[3]};
    v4f b2 = {xv[4], xv[5], xv[6], xv[7]};
    *(v4f*)&xtA[ml * LDS_STRIDE + 8 * hi] = a;
    *(v4f*)&xtA[ml * LDS_STRIDE + 8 * hi + 4] = b2;
    __syncthreads();
    v2f XB[4];
    lds_read_xb(xtA, XB);
    v8f y = do_step(XB);
#pragma unroll
    for (int j = 0; j < 8; ++j) {
      const int m = j + 8 * hi;
      const int jj = 16 * blk + m;
      const int i = dir ? (ELEN - 1 - jj) : jj;
      if (dir) {
        if (i >= PAD && i < PAD + TT) out[rbase + i - PAD] = y[j];
      } else {
        write_fwd(out, edge, rbase, redge, i, y[j]);
      }
    }
  };

#if HAVE_TDM
  // TDM descriptor per ISA ch.8: 2D tile, 16(time) x 16(rows), row stride TT
  // elements, 4B data, LDS pad 4 DWORDs every 16 DWORDs -> LDS_STRIDE = 20.
  auto issue_tdm = [&](int blk, float* buf) {
    const float* src = x + tilebase + (size_t)(16 * blk - PAD);
    unsigned long long ga = (unsigned long long)src;
    unsigned ldsoff = (unsigned)(unsigned long long)buf;  // addr[31:0] = LDS byte offset
    u32x4 g0;
    g0[0] = 1u;                                           // count=1, user desc
    g0[1] = ldsoff;                                       // lds_addr
    g0[2] = (unsigned)(ga & 0xffffffffull);               // global_addr[31:0]
    g0[3] = (unsigned)((ga >> 32) & 0x01ffffffull) | (2u << 30);  // [56:32]|type=2
    i32x8 g1;
    g1[0] = (int)((2u << 16)        // data_size = 4 bytes
                  | (1u << 20)      // pad_enable
                  | (3u << 22)      // pad_interval = 16 DWORDs
                  | (3u << 25));    // pad_amount   = 4 DWORDs
    g1[1] = (int)(16u << 16);       // tensor_dim0 = 16   (bits 79:48 low half)
    g1[2] = (int)(16u << 16);       // tensor_dim1 = 16   (bits 111:80 low half)
    g1[3] = (int)(16u << 16);       // tile_dim0 = 16     (bits 127:112)
    g1[4] = 16;                     // tile_dim1 = 16, tile_dim2 = 0
    g1[5] = TT;                     // tensor_dim0_stride[31:0] = 16384 elems
    g1[6] = 0;                      // stride0 hi, stride1 lo
    g1[7] = 0;                      // stride1 hi
    i32x4 gz = {0, 0, 0, 0};
    i32x8 gz8 = {0, 0, 0, 0, 0, 0, 0, 0};
    // 6-arg form (clang-23 / therock-10.0 headers): groups 2/3 + extra group
    // zero-filled (2D tensor), cpol = 0.
    __builtin_amdgcn_tensor_load_to_lds(g0, g1, gz, gz, gz8, 0);
  };
#endif

  auto load_tile_fast_fwd = [&](int blk, float* buf) {
    const float* src = x + rbase + (16 * blk - PAD) + 8 * hi;
    v4f a = {src[0], src[1], src[2], src[3]};
    v4f b2 = {src[4], src[5], src[6], src[7]};
    *(v4f*)&buf[ml * LDS_STRIDE + 8 * hi] = a;
    *(v4f*)&buf[ml * LDS_STRIDE + 8 * hi + 4] = b2;
  };

  // ------------------------------ forward ------------------------------
  {
    const float f0 = read_fwd(x, rbase, 0);
    SB[0].x = zi_a * f0; SB[0].y = zi_b * f0;
    SB[1].x = zi_c * f0; SB[1].y = zi_d * f0;

    edge_block(0, 0);
    edge_block(0, 1);
#if HAVE_TDM
    issue_tdm(2, xtA);
    for (int blk = 2; blk <= 1024; ++blk) {
      float* cur = (blk & 1) ? xtB : xtA;
      if (blk < 1024) {
        issue_tdm(blk + 1, (blk & 1) ? xtA : xtB);
        __builtin_amdgcn_s_wait_tensorcnt(1);  // current tile complete
      } else {
        __builtin_amdgcn_s_wait_tensorcnt(0);
      }
      __syncthreads();
      v2f XB[4];
      lds_read_xb(cur, XB);
      v8f y = do_step(XB);
#pragma unroll
      for (int j = 0; j < 8; ++j)
        out[rbase + 16 * blk + (j + 8 * hi) - PAD] = y[j];
    }
#else
    for (int blk = 2; blk <= 1024; ++blk) {
      load_tile_fast_fwd(blk, xtA);
      __syncthreads();
      v2f XB[4];
      lds_read_xb(xtA, XB);
      v8f y = do_step(XB);
#pragma unroll
      for (int j = 0; j < 8; ++j)
        out[rbase + 16 * blk + (j + 8 * hi) - PAD] = y[j];
    }
#endif
    edge_block(0, 1025);
    edge_block(0, 1026);
    edge_block(0, 1027);
  }

  // ------------------------------ backward -----------------------------
  {
    const float f0 = read_bwd(out, edge, rbase, redge, ELEN - 1);
    SB[0].x = zi_a * f0; SB[0].y = zi_b * f0;
    SB[1].x = zi_c * f0; SB[1].y = zi_d * f0;

    edge_block(1, 0);
    edge_block(1, 1);
    for (int bs = 2; bs <= 1024; ++bs) {
      // bulk reversed tile: all indices inside the main region of `out`
      const int tbase = (ELEN - 1 - PAD) - 16 * bs - 8 * hi;
      float xv[8];
#pragma unroll
      for (int q = 0; q < 8; ++q) xv[q] = out[rbase + tbase - q];
      v4f a = {xv[0], xv[1], xv[2], xv[3]};
      v4f b2 = {xv[4], xv[5], xv[6], xv[7]};
      *(v4f*)&xtA[ml * LDS_STRIDE + 8 * hi] = a;
      *(v4f*)&xtA[ml * LDS_STRIDE + 8 * hi + 4] = b2;
      __syncthreads();
      v2f XB[4];
      lds_read_xb(xtA, XB);
      v8f y = do_step(XB);
#pragma unroll
      for (int j = 0; j < 8; ++j)
        out[rbase + (ELEN - 1 - PAD) - (16 * bs + j + 8 * hi)] = y[j];
    }
    edge_block(1, 1025);
    edge_block(1, 1026);
    edge_block(1, 1027);
  }
}

// ---------------------------------------------------------------------------
extern "C" void kernel_launch(void* const* d_in, const int* in_sizes, int n_in,
                              void* d_out, int out_size, void* d_ws,
                              size_t ws_size, hipStream_t stream) {
  const float* x = (const float*)d_in[0];
  const float* b = (const float*)d_in[1];
  const float* a = (const float*)d_in[2];
  const float* zi = (const float*)d_in[3];
  float* out = (float*)d_out;
  float* ws = (float*)d_ws;
  float* mats = ws;            // 4 * 256 floats
  float* edgebuf = ws + 1024;  // BATCH * EDGE_STRIDE floats (256 KB)

  gen_mats_kernel<<<1, 1, 0, stream>>>(b, a, mats);
  filtfilt_wmma_kernel<<<BATCH / 16, 32, 0, stream>>>(x, zi, mats, edgebuf,
                                                      out);
}